// MultiDimensionalRNN_79413945303583
// MI455X (gfx1250) — compile-verified
//
#include <hip/hip_runtime.h>
#include <math.h>

// MI455X / gfx1250, wave32. Fused MDRNN + FC.
//  - x tile (16 batches x 1024 f32) staged global->LDS with CDNA5 async copy
//    (global_load_async_to_lds_b128, ASYNCcnt), no VGPR round-trip.
//  - one wave per batch element, lane = row, __shfl_up provides hid[t-1];
//    scan runs in-place in LDS (input slot consumed the same step its act
//    is produced).
//  - FC layer via V_WMMA_F32_16X16X4_F32 (full fp32): M=16 batches, N=16
//    (10 classes zero-padded), K=1024 split 64 per wave. A/B fragments are
//    preloaded so the 16 WMMAs issue as an uninterrupted accumulation chain.
//  - cross-wave C reduction through the reused LDS tile.

typedef __attribute__((ext_vector_type(2))) float v2f;
typedef __attribute__((ext_vector_type(8))) float v8f;

static constexpr int Hh     = 32;
static constexpr int Ww     = 32;
static constexpr int NC     = 10;
static constexpr int KTOT   = Hh * Ww;        // 1024
static constexpr int BPB    = 16;             // batches per block (= WMMA M)
static constexpr int NWAVE  = 16;             // 512 threads
static constexpr int KCHUNK = KTOT / NWAVE;   // 64 per wave
static constexpr int NWMMA  = KCHUNK / 4;     // 16 WMMA ops per wave

__global__ __launch_bounds__(512) void mdrnn_fused(
    const float* __restrict__ x,
    const float* __restrict__ w_state,
    const float* __restrict__ b_state_p,
    const float* __restrict__ w_in_p,
    const float* __restrict__ b_in_p,
    const float* __restrict__ fc_w,
    const float* __restrict__ fc_b,
    float* __restrict__ out)
{
    __shared__ float buf[BPB * KTOT] __attribute__((aligned(16))); // 64 KB

    const int tid  = threadIdx.x;
    const int wave = tid >> 5;
    const int lane = tid & 31;
    const int b0   = blockIdx.x * BPB;

    // ---- stage x tile: 4096 x 16B, async global->LDS (ASYNCcnt path) ----
    {
        const float4* xg   = (const float4*)(x + (size_t)b0 * KTOT);
        float4*       bs4  = (float4*)buf;
        #pragma unroll
        for (int i = 0; i < (BPB * KTOT / 4) / 512; ++i) {
            const int idx = tid + i * 512;
            const float4* g = xg + idx;
            // low 32 bits of a generic LDS pointer == LDS byte offset
            unsigned lo = (unsigned)(uintptr_t)(bs4 + idx);
            asm volatile("global_load_async_to_lds_b128 %0, %1, off"
                         :: "v"(lo), "v"(g) : "memory");
        }
        asm volatile("s_wait_asynccnt 0x0" ::: "memory");
    }
    __syncthreads();

    const float w0  = w_state[0];
    const float w1  = w_state[1];
    const float bst = b_state_p[0];
    const float wi  = w_in_p[0];
    const float bi  = b_in_p[0];

    // ---- phase 1: diagonal scan. wave -> batch b0+wave, lane -> row t ----
    {
        float* row = buf + wave * KTOT;
        const int t = lane;
        float hid = 0.0f;
        for (int j = 0; j < Hh + Ww - 1; ++j) {
            const int  cw    = j - t;
            const bool valid = (cw >= 0) && (cw < Ww);
            // skewed[b,t,j] = x[b,t,j-t] if valid else 0 (padding)
            float xv  = valid ? row[t * Ww + cw] : 0.0f;
            float inv = wi * xv + bi;                 // 1x1 conv affine, incl. pad
            float hprev = __shfl_up(hid, 1, 32);      // hid[t-1]
            if (t == 0) hprev = 0.0f;
            float s = w0 * hprev + w1 * hid + bst + inv;
            hid = tanhf(s);
            if (valid) row[t * Ww + cw] = hid;        // in-place overwrite
        }
    }
    __syncthreads();

    // ---- phase 2: WMMA f32 16x16x4, wave w covers K in [w*64, w*64+64) ----
    // A (16x4 f32): lanes 0-15 M=lane hold K pair {0,1}; lanes 16-31 M=lane-16
    // hold K pair {2,3}; VGPR index selects K within the pair.
    const int lm   = lane & 15;
    const int half = lane >> 4;
    const int kbase = wave * KCHUNK;

    const float* fcw_row = fc_w + (size_t)(lm < NC ? lm : (NC - 1)) * KTOT;
    const v2f    vzero   = {0.0f, 0.0f};

    v2f afrag[NWMMA], bfrag[NWMMA];
    #pragma unroll
    for (int s = 0; s < NWMMA; ++s) {
        const int k0 = kbase + s * 4 + half * 2;
        afrag[s] = *(const v2f*)(buf + lm * KTOT + k0);   // A[m=lm][k0..k0+1]
        v2f t    = *(const v2f*)(fcw_row + k0);           // B[k][n] = fc_w[n][k]
        bfrag[s] = (lm < NC) ? t : vzero;                 // branchless pad N 10->16
    }

    v8f c = {};
    #pragma unroll
    for (int s = 0; s < NWMMA; ++s) {
        c = __builtin_amdgcn_wmma_f32_16x16x4_f32(
                /*neg_a=*/false, afrag[s], /*neg_b=*/false, bfrag[s],
                /*c_mod=*/(short)0, c, /*reuse_a=*/false, /*reuse_b=*/false);
    }
    __syncthreads();   // all A reads from buf done -> safe to reuse buf

    // store per-wave partial C (16x16) into reused LDS
    #pragma unroll
    for (int i = 0; i < 8; ++i)
        buf[wave * 256 + i * 32 + lane] = c[i];
    __syncthreads();

    // reduce 16 partials; C layout: VGPR i, lane<16 -> (M=i, N=lane),
    // lane>=16 -> (M=8+i, N=lane-16)
    if (tid < 256) {
        float sum = 0.0f;
        #pragma unroll
        for (int w = 0; w < NWAVE; ++w)
            sum += buf[w * 256 + tid];
        const int i = tid >> 5;
        const int L = tid & 31;
        const int m = (L < 16) ? i : (i + 8);
        const int n = L & 15;
        if (n < NC)
            out[(size_t)(b0 + m) * NC + n] = sum + fc_b[n];
    }
}

extern "C" void kernel_launch(void* const* d_in, const int* in_sizes, int n_in,
                              void* d_out, int out_size, void* d_ws, size_t ws_size,
                              hipStream_t stream) {
    const float* x       = (const float*)d_in[0];
    const float* w_state = (const float*)d_in[1];
    const float* b_state = (const float*)d_in[2];
    const float* w_in    = (const float*)d_in[3];
    const float* b_in    = (const float*)d_in[4];
    const float* fc_w    = (const float*)d_in[5];
    const float* fc_b    = (const float*)d_in[6];
    float* out = (float*)d_out;

    const int B = in_sizes[0] / KTOT;      // 16384
    const int blocks = B / BPB;            // 1024
    mdrnn_fused<<<blocks, 512, 0, stream>>>(x, w_state, b_state, w_in, b_in,
                                            fc_w, fc_b, out);
}